// GAT_Classifier_90821378441672
// MI455X (gfx1250) — compile-verified
//
#include <hip/hip_runtime.h>
#include <math.h>

// Problem constants (from reference)
constexpr int NN   = 50000;   // nodes
constexpr int EE   = 400000;  // edges
constexpr int FIN  = 89;      // input features
constexpr int HH   = 6;       // heads
constexpr int DD   = 32;      // dim per head
constexpr int HD   = HH * DD; // 192
constexpr int GG   = 10;      // graphs
constexpr int CC   = 10;      // classes

typedef __attribute__((ext_vector_type(2))) float v2f;
typedef __attribute__((ext_vector_type(8))) float v8f;

__device__ inline v8f wmma_f32_k4(v2f a, v2f b, v8f c) {
  return __builtin_amdgcn_wmma_f32_16x16x4_f32(false, a, false, b, (short)0, c,
                                               false, false);
}

// ---------------------------------------------------------------------------
// fp32 WMMA GEMM: C[NN,192] = A[NN,K] @ B[K,192].
// Block = 96 threads (3 waves) <-> one 16-row strip of A staged in LDS
// (zero-padded to KP, row stride KP+2 to spread banks, 8B aligned).
// Each wave owns 4 consecutive 16-col tiles (64 cols), reusing the A fragment
// across 4 WMMAs per K-step. Main loop is guard-free; K%4 tail is a
// compile-time-separated single step (only instantiated for K=89).
// A-fragment layout (ISA 7.12.2): lanes 0-15 hold K=k,k+1 of row M=lane&15;
// lanes 16-31 hold K=k+2,k+3.  D: acc[r] = (M = r + 8*(lane>=16), N = lane&15).
// ---------------------------------------------------------------------------
template <int K>
__global__ __launch_bounds__(96) void k_gemm_wmma(const float* __restrict__ A,
                                                  const float* __restrict__ B,
                                                  float* __restrict__ Cout) {
  constexpr int KP    = (K + 3) & ~3;   // zero-padded K (92 or 192)
  constexpr int KMAIN = K & ~3;         // guard-free region (88 or 192)
  constexpr int LDK   = KP + 2;         // LDS row stride in floats (even)
  __shared__ float As[16 * LDK];

  const int tid  = threadIdx.x;
  const int row0 = blockIdx.x << 4;

  // Cooperative stage of the 16xK A panel, zero-padded to KP.
  for (int i = tid; i < 16 * KP; i += 96) {
    int r = i / KP, k = i - r * KP;
    As[r * LDK + k] = (k < K) ? A[(size_t)(row0 + r) * K + k] : 0.0f;
  }
  __syncthreads();

  const int wid  = tid >> 5;            // 0..2
  const int lane = tid & 31;
  const int lh   = lane >> 4;           // half select
  const int lm   = lane & 15;
  const int col0 = wid * 64;

  v8f acc0 = {}, acc1 = {}, acc2 = {}, acc3 = {};
  const float* as = &As[lm * LDK + 2 * lh];
  const float* bp = B + (size_t)(2 * lh) * HD + col0 + lm;

  for (int k = 0; k < KMAIN; k += 4, bp += 4 * (size_t)HD) {
    v2f a = *(const v2f*)(as + k);      // ds_load_b64, no guards
    const float* b0 = bp;               // W row k+2*lh
    const float* b1 = bp + HD;          // W row k+2*lh+1
    v2f b;
    b.x = b0[0];  b.y = b1[0];  acc0 = wmma_f32_k4(a, b, acc0);
    b.x = b0[16]; b.y = b1[16]; acc1 = wmma_f32_k4(a, b, acc1);
    b.x = b0[32]; b.y = b1[32]; acc2 = wmma_f32_k4(a, b, acc2);
    b.x = b0[48]; b.y = b1[48]; acc3 = wmma_f32_k4(a, b, acc3);
  }

  if (K != KMAIN) {                     // compile-time: only for K=89
    const int ka = KMAIN + 2 * lh;
    v2f a = *(const v2f*)(as + KMAIN);  // LDS zero-padded -> safe
    const bool v0 = (ka < K), v1 = (ka + 1 < K);
    const float* b0 = B + (size_t)ka * HD + col0 + lm;
    const float* b1 = b0 + HD;
    v2f b;
    b.x = v0 ? b0[0]  : 0.0f; b.y = v1 ? b1[0]  : 0.0f; acc0 = wmma_f32_k4(a, b, acc0);
    b.x = v0 ? b0[16] : 0.0f; b.y = v1 ? b1[16] : 0.0f; acc1 = wmma_f32_k4(a, b, acc1);
    b.x = v0 ? b0[32] : 0.0f; b.y = v1 ? b1[32] : 0.0f; acc2 = wmma_f32_k4(a, b, acc2);
    b.x = v0 ? b0[48] : 0.0f; b.y = v1 ? b1[48] : 0.0f; acc3 = wmma_f32_k4(a, b, acc3);
  }

#pragma unroll
  for (int r = 0; r < 8; ++r) {
    float* cr = Cout + (size_t)(row0 + r + 8 * lh) * HD + col0 + lm;
    cr[0] = acc0[r]; cr[16] = acc1[r]; cr[32] = acc2[r]; cr[48] = acc3[r];
  }
}

// el[n,h] = sum_d feat[n,h,d]*al[h,d];  er likewise
__global__ void k_attn_logits(const float* __restrict__ feat,
                              const float* __restrict__ al, const float* __restrict__ ar,
                              float* __restrict__ el, float* __restrict__ er) {
  int idx = blockIdx.x * blockDim.x + threadIdx.x;  // n*H + h
  if (idx >= NN * HH) return;
  int n = idx / HH, h = idx % HH;
  const float* f  = feat + (size_t)n * HD + h * DD;
  const float* a1 = al + h * DD;
  const float* a2 = ar + h * DD;
  float s1 = 0.f, s2 = 0.f;
#pragma unroll
  for (int d = 0; d < DD; ++d) { float v = f[d]; s1 += v * a1[d]; s2 += v * a2[d]; }
  el[idx] = s1; er[idx] = s2;
}

__global__ void k_init_attn(float* __restrict__ emax, float* __restrict__ denom) {
  int i = blockIdx.x * blockDim.x + threadIdx.x;
  if (i < NN * HH) { emax[i] = -INFINITY; denom[i] = 0.0f; }
}

__global__ void k_init_rst(float* __restrict__ rst, const float* __restrict__ bias) {
  int i = blockIdx.x * blockDim.x + threadIdx.x;
  if (i < NN * HD) rst[i] = bias[i % HD];
}

__device__ inline void atomicMaxF(float* addr, float val) {
  // punning trick: IEEE ordering preserved for any sign
  if (val >= 0.0f) atomicMax((int*)addr, __float_as_int(val));
  else             atomicMin((unsigned int*)addr, __float_as_uint(val));
}

// e = leaky_relu(el[src]+er[dst]); emax[dst,h] = max(...)
__global__ void k_edge_logits(const float* __restrict__ el, const float* __restrict__ er,
                              const int* __restrict__ src, const int* __restrict__ dst,
                              float* __restrict__ e, float* __restrict__ emax) {
  int idx = blockIdx.x * blockDim.x + threadIdx.x;  // edge*H + h
  if (idx >= EE * HH) return;
  int ed = idx / HH, h = idx % HH;
  int s = src[ed], d = dst[ed];
  float v = el[s * HH + h] + er[d * HH + h];
  v = (v > 0.0f) ? v : 0.2f * v;
  e[idx] = v;
  atomicMaxF(&emax[d * HH + h], v);
}

// e <- exp(e - emax[dst]); denom[dst,h] += e
__global__ void k_edge_exp(float* __restrict__ e, const float* __restrict__ emax,
                           const int* __restrict__ dst, float* __restrict__ denom) {
  int idx = blockIdx.x * blockDim.x + threadIdx.x;
  if (idx >= EE * HH) return;
  int ed = idx / HH, h = idx % HH;
  int d = dst[ed];
  float ee = __expf(e[idx] - emax[d * HH + h]);
  e[idx] = ee;
  atomicAdd(&denom[d * HH + h], ee);
}

// One wave per edge, lane = D index. rst[dst,h,:] += feat[src,h,:] * alpha_h
__global__ void k_edge_aggregate(const float* __restrict__ feat, const float* __restrict__ e,
                                 const float* __restrict__ denom,
                                 const int* __restrict__ src, const int* __restrict__ dst,
                                 float* __restrict__ rst) {
  int wave = (int)((blockIdx.x * blockDim.x + threadIdx.x) >> 5);
  int lane = threadIdx.x & 31;
  if (wave >= EE) return;
  int s = src[wave], d = dst[wave];
  const float* fs = feat + (size_t)s * HD;
  float* rd = rst + (size_t)d * HD;
#pragma unroll
  for (int h = 0; h < HH; ++h) {
    float alpha = e[(size_t)wave * HH + h] / denom[d * HH + h];
    atomicAdd(&rd[h * DD + lane], fs[h * DD + lane] * alpha);
  }
}

__global__ void k_init_pool(float* __restrict__ gsum, float* __restrict__ gcnt) {
  int i = blockIdx.x * blockDim.x + threadIdx.x;
  if (i < GG * HD) gsum[i] = 0.0f;
  if (i < GG)      gcnt[i] = 0.0f;
}

__global__ void k_pool(const float* __restrict__ hfin, const int* __restrict__ gid,
                       float* __restrict__ gsum, float* __restrict__ gcnt) {
  int idx = blockIdx.x * blockDim.x + threadIdx.x;  // n*HD + j
  if (idx >= NN * HD) return;
  int n = idx / HD, j = idx % HD;
  int g = gid[n];
  atomicAdd(&gsum[g * HD + j], hfin[idx]);
  if (j == 0) atomicAdd(&gcnt[g], 1.0f);
}

// Single block: hg = gsum/max(cnt,1); out = relu(hg@Wc1+bc1)@Wc2+bc2
__global__ void k_classifier(const float* __restrict__ gsum, const float* __restrict__ gcnt,
                             const float* __restrict__ Wc1, const float* __restrict__ bc1,
                             const float* __restrict__ Wc2, const float* __restrict__ bc2,
                             float* __restrict__ out) {
  __shared__ float hg[GG * HD];
  __shared__ float z[GG * DD];
  for (int i = threadIdx.x; i < GG * HD; i += blockDim.x) {
    int g = i / HD;
    float c = gcnt[g];
    c = (c > 1.0f) ? c : 1.0f;
    hg[i] = gsum[i] / c;
  }
  __syncthreads();
  for (int i = threadIdx.x; i < GG * DD; i += blockDim.x) {
    int g = i / DD, d = i % DD;
    float sacc = bc1[d];
    for (int k = 0; k < HD; ++k) sacc += hg[g * HD + k] * Wc1[k * DD + d];
    z[i] = (sacc > 0.0f) ? sacc : 0.0f;
  }
  __syncthreads();
  for (int i = threadIdx.x; i < GG * CC; i += blockDim.x) {
    int g = i / CC, c = i % CC;
    float sacc = bc2[c];
    for (int k = 0; k < DD; ++k) sacc += z[g * DD + k] * Wc2[k * CC + c];
    out[i] = sacc;
  }
}

static inline int cdiv(long long a, int b) { return (int)((a + b - 1) / b); }

extern "C" void kernel_launch(void* const* d_in, const int* in_sizes, int n_in,
                              void* d_out, int out_size, void* d_ws, size_t ws_size,
                              hipStream_t stream) {
  const float* nf    = (const float*)d_in[0];
  const float* W[3]  = {(const float*)d_in[1], (const float*)d_in[5], (const float*)d_in[9]};
  const float* al[3] = {(const float*)d_in[2], (const float*)d_in[6], (const float*)d_in[10]};
  const float* ar[3] = {(const float*)d_in[3], (const float*)d_in[7], (const float*)d_in[11]};
  const float* bb[3] = {(const float*)d_in[4], (const float*)d_in[8], (const float*)d_in[12]};
  const float* Wc1 = (const float*)d_in[13];
  const float* bc1 = (const float*)d_in[14];
  const float* Wc2 = (const float*)d_in[15];
  const float* bc2 = (const float*)d_in[16];
  const int*   src = (const int*)d_in[17];
  const int*   dst = (const int*)d_in[18];
  const int*   gid = (const int*)d_in[19];
  float* out = (float*)d_out;

  // Workspace layout (floats): ~129.6 MB total
  float* ws = (float*)d_ws;
  size_t off = 0;
  float* feat  = ws + off; off += (size_t)NN * HD;   // 9.6 M
  float* hA    = ws + off; off += (size_t)NN * HD;   // 9.6 M
  float* hB    = ws + off; off += (size_t)NN * HD;   // 9.6 M
  float* elb   = ws + off; off += (size_t)NN * HH;   // 0.3 M
  float* erb   = ws + off; off += (size_t)NN * HH;
  float* emax  = ws + off; off += (size_t)NN * HH;
  float* denom = ws + off; off += (size_t)NN * HH;
  float* eedge = ws + off; off += (size_t)EE * HH;   // 2.4 M
  float* gsum  = ws + off; off += (size_t)GG * HD;
  float* gcnt  = ws + off; off += (size_t)GG;

  const float* hin[3]  = {nf, hA, hB};
  float*       hout[3] = {hA, hB, hA};

  const int TPB        = 256;
  const int gemm_grid  = NN / 16;                         // 3125 row strips
  const int nh_blocks  = cdiv((long long)NN * HH, TPB);
  const int nhd_blocks = cdiv((long long)NN * HD, TPB);
  const int eh_blocks  = cdiv((long long)EE * HH, TPB);
  const int agg_blocks = cdiv((long long)EE * 32, TPB);   // wave per edge

  for (int L = 0; L < 3; ++L) {
    // feat = hin @ W[L]   (fp32 WMMA, K specialized at compile time)
    if (L == 0) k_gemm_wmma<FIN><<<gemm_grid, 96, 0, stream>>>(hin[L], W[L], feat);
    else        k_gemm_wmma<HD> <<<gemm_grid, 96, 0, stream>>>(hin[L], W[L], feat);
    // attention logits per node/head
    k_attn_logits<<<nh_blocks, TPB, 0, stream>>>(feat, al[L], ar[L], elb, erb);
    // init segment-softmax state and output accumulator (bias)
    k_init_attn<<<nh_blocks, TPB, 0, stream>>>(emax, denom);
    k_init_rst<<<nhd_blocks, TPB, 0, stream>>>(hout[L], bb[L]);
    // edge softmax
    k_edge_logits<<<eh_blocks, TPB, 0, stream>>>(elb, erb, src, dst, eedge, emax);
    k_edge_exp<<<eh_blocks, TPB, 0, stream>>>(eedge, emax, dst, denom);
    // weighted aggregation
    k_edge_aggregate<<<agg_blocks, TPB, 0, stream>>>(feat, eedge, denom, src, dst, hout[L]);
  }

  // mean pooling per graph + classifier MLP
  k_init_pool<<<cdiv(GG * HD, TPB), TPB, 0, stream>>>(gsum, gcnt);
  k_pool<<<nhd_blocks, TPB, 0, stream>>>(hA, gid, gsum, gcnt);  // hout[2] == hA
  k_classifier<<<1, TPB, 0, stream>>>(gsum, gcnt, Wc1, bc1, Wc2, bc2, out);

  (void)in_sizes; (void)n_in; (void)out_size; (void)ws_size;
}